// EdgeTransformer_41068477284525
// MI455X (gfx1250) — compile-verified
//
#include <hip/hip_runtime.h>
#include <hip/hip_bf16.h>

// ---------------- problem constants (match reference) ----------------
#define BB   2
#define NN   64
#define DD   128
#define HH   8
#define DK   16
#define EE   2048
#define TOK  (BB * NN * NN)          // 8192 pair tokens
#define NODES (BB * NN)              // 128 nodes
#define MAXK 256                     // largest GEMM K in this model

typedef float v2f __attribute__((ext_vector_type(2)));
typedef float v8f __attribute__((ext_vector_type(8)));
typedef int   v2i __attribute__((ext_vector_type(2)));

// f32 WMMA: D(16x16,f32) = A(16x4,f32) * B(4x16,f32) + C
__device__ __forceinline__ v8f wmma_f32_4(v2f a, v2f b, v8f c) {
  return __builtin_amdgcn_wmma_f32_16x16x4_f32(false, a, false, b, (short)0, c,
                                               false, false);
}

// ---- CDNA5 async global->LDS copy (ASYNCcnt) with safe fallback ----
#if defined(__has_builtin)
#if __has_builtin(__builtin_amdgcn_global_load_async_to_lds_b64) && \
    __has_builtin(__builtin_amdgcn_s_wait_asynccnt)
#define HAS_ASYNC_LDS 1
#endif
#endif

#ifdef HAS_ASYNC_LDS
typedef __attribute__((address_space(1))) v2i  g_v2i;   // global int2*
typedef __attribute__((address_space(3))) v2i  l_v2i;   // LDS int2*
typedef __attribute__((address_space(3))) void l_void;
__device__ __forceinline__ void async_cp_b64(const float* g, float* l) {
  g_v2i* gp = (g_v2i*)(unsigned long long)g;   // int -> AS1 pointer
  l_void* lv = (l_void*)l;                     // generic -> AS3 (addrspacecast)
  l_v2i* lp = (l_v2i*)lv;                      // AS3 void* -> AS3 int2*
  __builtin_amdgcn_global_load_async_to_lds_b64(gp, lp, 0, 0);
}
#endif

// ---------------------------------------------------------------------
// Generic fp32 WMMA GEMM: C[M,N] = A[M,K] * W[K,N] (+bias) (+res) (ReLU?)
// Block = 4 waves sharing one 16-row A panel staged in LDS (async copy);
// each wave owns one 16-col N tile.  M,N multiples of 16; K mult of 4,
// K <= MAXK.
// ---------------------------------------------------------------------
__global__ void gemm16(const float* __restrict__ A, const float* __restrict__ W,
                       const float* __restrict__ bias,
                       const float* __restrict__ res, float* __restrict__ C,
                       int M, int Nn, int K, int relu) {
  __shared__ float sA[16 * MAXK];              // 16 KB max

  int tm = blockIdx.x << 4;                    // M tile (rows tm..tm+15)
  int wave = threadIdx.x >> 5;                 // 0..3
  int lane = threadIdx.x & 31;
  int lm = lane & 15;
  int hi = lane >> 4;
  int kk = hi << 1;

  // ---- cooperative copy of A[tm..tm+15][0..K) into LDS ----
  const float* Ablk = A + (size_t)tm * K;
  int elems = 16 * K;                          // multiple of 256
#ifdef HAS_ASYNC_LDS
  for (int i = threadIdx.x * 2; i < elems; i += blockDim.x * 2)
    async_cp_b64(Ablk + i, &sA[i]);
  __builtin_amdgcn_s_wait_asynccnt(0);
#else
  for (int i = threadIdx.x * 2; i < elems; i += blockDim.x * 2)
    *(v2f*)&sA[i] = *(const v2f*)(Ablk + i);
#endif
  __syncthreads();

  int tn = (blockIdx.y * 4 + wave) << 4;       // this wave's N tile
  if (tn >= Nn) return;                        // wave-uniform, after the only barrier

  const float* bcol = W + (size_t)kk * Nn + tn + lm;   // W[kk+v][n]
  __builtin_prefetch(bcol, 0, 1);              // global_prefetch_b8 (weights panel)

  v8f acc = {0.f, 0.f, 0.f, 0.f, 0.f, 0.f, 0.f, 0.f};
  const float* arow = &sA[lm * K + kk];        // ds_load_b64 per k-step

  for (int k0 = 0; k0 < K; k0 += 4) {
    v2f af = *(const v2f*)(arow + k0);
    v2f bf;
    bf.x = bcol[(size_t)k0 * Nn];
    bf.y = bcol[(size_t)(k0 + 1) * Nn];
    acc = wmma_f32_4(af, bf, acc);
  }

#pragma unroll
  for (int r = 0; r < 8; ++r) {
    int m = tm + r + hi * 8;
    int n = tn + lm;
    float v = acc[r];
    if (bias) v += bias[n];
    if (res) v += res[(size_t)m * Nn + n];
    if (relu) v = fmaxf(v, 0.f);
    C[(size_t)m * Nn + n] = v;
  }
}

// ---------------------------------------------------------------------
// Composer: build P[t] = [x_i ; x_j]  (token_index is row-major all pairs)
// ---------------------------------------------------------------------
__global__ void pair_build(const float* __restrict__ x, float* __restrict__ P) {
  int t = blockIdx.x;                         // 0..TOK-1
  int b = t / (NN * NN);
  int r = t % (NN * NN);
  int i = r / NN, j = r % NN;
  int c = threadIdx.x;                        // 0..255
  float v = (c < DD) ? x[(size_t)(b * NN + i) * DD + c]
                     : x[(size_t)(b * NN + j) * DD + (c - DD)];
  P[(size_t)t * (2 * DD) + c] = v;
}

// dense_edge scatter-add of edge_emb row 1 (edges may repeat -> atomics)
__global__ void edge_add(const int* __restrict__ ei, const float* __restrict__ emb,
                         float* __restrict__ X) {
  int e = blockIdx.x;
  int src = ei[e], dst = ei[EE + e];
  int b = src / NN;
  int ii = src - b * NN, jj = dst - b * NN;
  atomicAdd(&X[((size_t)(b * NN + ii) * NN + jj) * DD + threadIdx.x],
            emb[DD + threadIdx.x]);
}

// ---------------------------------------------------------------------
// LayerNorm over D=128, optional elementwise addend (residual), wave32.
// ---------------------------------------------------------------------
__global__ void layernorm128(const float* __restrict__ X,
                             const float* __restrict__ addend,
                             const float* __restrict__ g,
                             const float* __restrict__ bsh,
                             float* __restrict__ out) {
  int row = blockIdx.x;
  int lane = threadIdx.x;                     // 32 lanes, 4 floats each
  float4 v = ((const float4*)(X + (size_t)row * DD))[lane];
  if (addend) {
    float4 a4 = ((const float4*)(addend + (size_t)row * DD))[lane];
    v.x += a4.x; v.y += a4.y; v.z += a4.z; v.w += a4.w;
  }
  float s = v.x + v.y + v.z + v.w;
  float s2 = v.x * v.x + v.y * v.y + v.z * v.z + v.w * v.w;
#pragma unroll
  for (int off = 16; off > 0; off >>= 1) {
    s += __shfl_xor(s, off, 32);
    s2 += __shfl_xor(s2, off, 32);
  }
  float mean = s * (1.f / DD);
  float var = s2 * (1.f / DD) - mean * mean;
  float rstd = rsqrtf(var + 1e-5f);
  float4 gv = ((const float4*)g)[lane];
  float4 bv = ((const float4*)bsh)[lane];
  float4 o;
  o.x = (v.x - mean) * rstd * gv.x + bv.x;
  o.y = (v.y - mean) * rstd * gv.y + bv.y;
  o.z = (v.z - mean) * rstd * gv.z + bv.z;
  o.w = (v.w - mean) * rstd * gv.w + bv.w;
  ((float4*)(out + (size_t)row * DD))[lane] = o;
}

// BatchNorm1d eval (running stats 0/1): y = x * g/sqrt(1+eps) + b
__global__ void scale_bias(const float* __restrict__ in, const float* __restrict__ g,
                           const float* __restrict__ b, float* __restrict__ out,
                           int total) {
  int i = blockIdx.x * blockDim.x + threadIdx.x;
  if (i >= total) return;
  int c = i & (DD - 1);
  out[i] = in[i] * (g[c] * 0.9999950000374997f) + b[c];
}

// ---------------------------------------------------------------------
// scores[b,h,x,a,y] = (1/sqrt(DK)) * sum_d lk[b,x,a,h,d]*rk[b,a,y,h,d]
// 1024 batched 64x64x16 WMMA GEMMs, one (b,h,a) per block, 16 waves.
// ---------------------------------------------------------------------
__global__ void scores_kernel(const float* __restrict__ lk,
                              const float* __restrict__ rk,
                              float* __restrict__ S) {
  int g = blockIdx.x;                         // (b*H + h)*N + a
  int a = g % NN;
  int h = (g / NN) % HH;
  int b = g / (NN * HH);
  int w = threadIdx.x >> 5;                   // 0..15
  int tm = (w >> 2) << 4;                     // x tile
  int tn = (w & 3) << 4;                      // y tile
  int lane = threadIdx.x & 31;
  int lm = lane & 15;
  int hi = lane >> 4;
  int kk = hi << 1;

  const float* Ab = lk + ((size_t)((b * NN + tm + lm) * NN + a)) * DD + h * DK + kk;
  const float* Bb = rk + ((size_t)((b * NN + a) * NN + tn + lm)) * DD + h * DK + kk;
  v8f acc = {0.f, 0.f, 0.f, 0.f, 0.f, 0.f, 0.f, 0.f};
#pragma unroll
  for (int k0 = 0; k0 < DK; k0 += 4) {
    v2f af = *(const v2f*)(Ab + k0);
    v2f bf = *(const v2f*)(Bb + k0);
    acc = wmma_f32_4(af, bf, acc);
  }
  size_t base = (((size_t)(b * HH + h) * NN + tm) * NN + a) * NN + tn;
#pragma unroll
  for (int r = 0; r < 8; ++r)
    S[base + (size_t)(r + hi * 8) * NN * NN + lm] = acc[r] * 0.25f;  // 1/sqrt(16)
}

// softmax over a (middle axis) for layout [bhx][a][y]; LDS-staged.
__global__ void softmax_a(float* __restrict__ S) {
  __shared__ float sm[NN * NN];
  size_t base = (size_t)blockIdx.x * NN * NN;
  for (int i = threadIdx.x; i < NN * NN; i += blockDim.x) sm[i] = S[base + i];
  __syncthreads();
  int y = threadIdx.x;                        // blockDim = 64
  float mx = -1e30f;
  for (int a = 0; a < NN; ++a) mx = fmaxf(mx, sm[a * NN + y]);
  float sum = 0.f;
  for (int a = 0; a < NN; ++a) sum += __expf(sm[a * NN + y] - mx);
  float inv = 1.f / sum;
  for (int a = 0; a < NN; ++a)
    S[base + a * NN + y] = __expf(sm[a * NN + y] - mx) * inv;
}

// o[b,x,y,h,d] = sum_a att[b,h,x,a,y] * lv[b,x,a,h,d] * rv[b,a,y,h,d]
__global__ void o_einsum(const float* __restrict__ ATT, const float* __restrict__ LV,
                         const float* __restrict__ RV, float* __restrict__ O) {
  __shared__ float s_att[NN * NN];            // 16 KB
  __shared__ float s_lv[NN * DK];             // 4 KB
  int g = blockIdx.x;                         // (b*H + h)*N + x
  int x = g % NN;
  int h = (g / NN) % HH;
  int b = g / (NN * HH);
  size_t abase = (size_t)g * NN * NN;
  for (int i = threadIdx.x; i < NN * NN; i += blockDim.x) s_att[i] = ATT[abase + i];
  for (int i = threadIdx.x; i < NN * DK; i += blockDim.x) {
    int a = i >> 4, d = i & 15;
    s_lv[i] = LV[((size_t)((b * NN + x) * NN + a)) * DD + h * DK + d];
  }
  __syncthreads();
  for (int o = threadIdx.x; o < NN * DK; o += blockDim.x) {
    int y = o >> 4, d = o & 15;
    float sum = 0.f;
    for (int a = 0; a < NN; ++a)
      sum += s_att[a * NN + y] * s_lv[(a << 4) + d] *
             RV[((size_t)((b * NN + a) * NN + y)) * DD + h * DK + d];
    O[((size_t)((b * NN + x) * NN + y)) * DD + h * DK + d] = sum;
  }
}

// node features: nf[u] = sum_j hd[u*N+j, :128] + sum_i hd[(bN+i)*N + u%N, 128:]
__global__ void segsum(const float* __restrict__ hd, float* __restrict__ nf) {
  int u = blockIdx.x;                         // 0..127
  int d = threadIdx.x;                        // 0..127
  int b = u / NN, jj = u % NN;
  float s = 0.f;
  for (int j = 0; j < NN; ++j) s += hd[((size_t)u * NN + j) * 256 + d];
  for (int i = 0; i < NN; ++i)
    s += hd[((size_t)((b * NN + i) * NN) + jj) * 256 + 128 + d];
  nf[(size_t)u * DD + d] = s;
}

// ---------------------------------------------------------------------
static inline void launch_gemm(const float* A, const float* W, const float* bias,
                               const float* res, float* C, int M, int Nn, int K,
                               int relu, hipStream_t st) {
  int ntn = Nn / 16;
  dim3 grid(M / 16, (ntn + 3) / 4);
  gemm16<<<grid, 128, 0, st>>>(A, W, bias, res, C, M, Nn, K, relu);
}

extern "C" void kernel_launch(void* const* d_in, const int* in_sizes, int n_in,
                              void* d_out, int out_size, void* d_ws, size_t ws_size,
                              hipStream_t stream) {
  (void)in_sizes; (void)n_in; (void)out_size; (void)ws_size;
  auto F = [&](int i) { return (const float*)d_in[i]; };
  const float* x = F(0);
  const int* edge_index = (const int*)d_in[1];
  // d_in[2]=node_batch, d_in[3]=token_index: row-major all pairs -> computed directly
  // params, insertion order:
  const int P_COMP_W = 4, P_COMP_B = 5, P_COMP_EMB = 6;
  const int L0 = 7;   // per layer (15): pre_g,pre_b,Wlk,Wrk,Wlv,Wrv,Wo,agg_g,agg_b,W1,b1,W2,b2,out_g,out_b
  const int PDEC = L0 + 30;   // bn1_g,bn1_b,W1,b1,W2,b2,bn2_g,bn2_b,Wn1,bn_1,Wn2,bn_2
  const int PHEAD = PDEC + 12;  // W1,b1,W2,b2,W3,b3

  // workspace layout (floats)
  float* ws = (float*)d_ws;
  float* X  = ws;                      // TOK*D
  float* XN = X + (size_t)TOK * DD;
  float* LK = XN + (size_t)TOK * DD;
  float* RK = LK + (size_t)TOK * DD;
  float* LV = RK + (size_t)TOK * DD;
  float* RV = LV + (size_t)TOK * DD;
  float* O  = RV + (size_t)TOK * DD;
  float* HB = O + (size_t)TOK * DD;
  float* ATT = HB + (size_t)TOK * DD;  // B*H*N*N*N = 4,194,304 floats
  float* T1 = ATT;                     // overlay: TOK*256 (composer pair / FFN / dec)
  float* T2 = ATT + (size_t)TOK * 256; // overlay: TOK*256 (dec hd)
  float* NF  = ATT + (size_t)BB * HH * NN * NN * NN;  // 128*128
  float* NF2 = NF + NODES * DD;
  float* HH1 = NF2 + NODES * DD;       // 128*64
  float* HH2 = HH1 + NODES * 64;       // 128*32

  // ---- composer: X = concat(x_i,x_j) @ W + b, then edge-embedding scatter ----
  pair_build<<<TOK, 256, 0, stream>>>(x, T1);
  launch_gemm(T1, F(P_COMP_W), F(P_COMP_B), nullptr, X, TOK, DD, 2 * DD, 0, stream);
  edge_add<<<EE, DD, 0, stream>>>(edge_index, F(P_COMP_EMB), X);

  // ---- transformer layers ----
  for (int l = 0; l < 2; ++l) {
    int p = L0 + 15 * l;
    layernorm128<<<TOK, 32, 0, stream>>>(X, nullptr, F(p + 0), F(p + 1), XN);
    launch_gemm(XN, F(p + 2), nullptr, nullptr, LK, TOK, DD, DD, 0, stream);
    launch_gemm(XN, F(p + 3), nullptr, nullptr, RK, TOK, DD, DD, 0, stream);
    launch_gemm(XN, F(p + 4), nullptr, nullptr, LV, TOK, DD, DD, 0, stream);
    launch_gemm(XN, F(p + 5), nullptr, nullptr, RV, TOK, DD, DD, 0, stream);
    scores_kernel<<<BB * HH * NN, 512, 0, stream>>>(LK, RK, ATT);
    softmax_a<<<BB * HH * NN, 64, 0, stream>>>(ATT);
    o_einsum<<<BB * HH * NN, 256, 0, stream>>>(ATT, LV, RV, O);
    launch_gemm(O, F(p + 6), nullptr, nullptr, XN, TOK, DD, DD, 0, stream); // x_upd
    layernorm128<<<TOK, 32, 0, stream>>>(X, XN, F(p + 7), F(p + 8), HB);    // h
    launch_gemm(HB, F(p + 9), F(p + 10), nullptr, T1, TOK, 2 * DD, DD, 1, stream);
    launch_gemm(T1, F(p + 11), F(p + 12), HB, XN, TOK, DD, 2 * DD, 0, stream);
    layernorm128<<<TOK, 32, 0, stream>>>(XN, nullptr, F(p + 13), F(p + 14), X);
  }

  // ---- decoder ----
  scale_bias<<<(TOK * DD) / 256, 256, 0, stream>>>(X, F(PDEC + 0), F(PDEC + 1), XN,
                                                   TOK * DD);
  launch_gemm(XN, F(PDEC + 2), F(PDEC + 3), nullptr, T1, TOK, 2 * DD, DD, 1, stream);
  launch_gemm(T1, F(PDEC + 4), F(PDEC + 5), nullptr, T2, TOK, 2 * DD, 2 * DD, 0, stream);
  segsum<<<NODES, DD, 0, stream>>>(T2, NF);
  scale_bias<<<(NODES * DD + 255) / 256, 256, 0, stream>>>(NF, F(PDEC + 6), F(PDEC + 7),
                                                           NF2, NODES * DD);
  launch_gemm(NF2, F(PDEC + 8), F(PDEC + 9), nullptr, NF, NODES, DD, DD, 1, stream);
  launch_gemm(NF, F(PDEC + 10), F(PDEC + 11), nullptr, NF2, NODES, DD, DD, 0, stream);

  // ---- head ----
  launch_gemm(NF2, F(PHEAD + 0), F(PHEAD + 1), nullptr, HH1, NODES, 64, DD, 1, stream);
  launch_gemm(HH1, F(PHEAD + 2), F(PHEAD + 3), nullptr, HH2, NODES, 32, 64, 1, stream);
  launch_gemm(HH2, F(PHEAD + 4), F(PHEAD + 5), nullptr, (float*)d_out, NODES, 32, 32,
              0, stream);
}